// Suppression_47536698032423
// MI455X (gfx1250) — compile-verified
//
#include <hip/hip_runtime.h>
#include <stdint.h>

#define NANCH 8732
#define NCLS  599
#define MM    300
#define KOUT  200
#define BATCH 2
#define NTOT  (NCLS*MM)          /* 179700 */
#define CONF_T 0.01f
#define IOU_T  0.45f
#define NEGV  (-1e30f)
#define KPT2  35                 /* 35*256 >= 8732 */

__device__ __forceinline__ uint32_t f2ord(float f) {
  uint32_t u = __float_as_uint(f);
  return (u >> 31) ? ~u : (u | 0x80000000u);
}
__device__ __forceinline__ float ord2f(uint32_t u) {
  uint32_t b = (u >> 31) ? (u ^ 0x80000000u) : ~u;
  return __uint_as_float(b);
}

// ---------------------------------------------------------------------------
// K1: transpose scores [B,N,604](col 1..599) -> scoresT [B,C,N], staged through
// LDS with gfx1250 async global->LDS copies (ASYNCcnt path).
// ---------------------------------------------------------------------------
__global__ __launch_bounds__(256) void k_transpose(const float* __restrict__ in,
                                                   float* __restrict__ scoresT) {
  __shared__ float tile[64][33];
  const int t  = threadIdx.x;
  const int n0 = blockIdx.x * 64;
  const int c0 = blockIdx.y * 32;
  const int b  = blockIdx.z;

  const int cl = t & 31;
  const int r  = t >> 5;                 // 0..7
#pragma unroll
  for (int s = 0; s < 8; ++s) {
    const int nl = r + 8 * s;            // 0..63
    const int n = n0 + nl, c = c0 + cl;
    if (n < NANCH && c < NCLS) {
      const float* g = in + ((size_t)b * NANCH + n) * 604 + 1 + c;
      uint32_t lds_off = (uint32_t)(uintptr_t)&tile[nl][cl];
      asm volatile("global_load_async_to_lds_b32 %0, %1, off"
                   :: "v"(lds_off), "v"(g) : "memory");
    }
  }
  asm volatile("s_wait_asynccnt 0" ::: "memory");
  __syncthreads();

  const int nl2 = t & 63;
  const int cq  = t >> 6;                // 0..3
#pragma unroll
  for (int s = 0; s < 8; ++s) {
    const int cl2 = cq + 4 * s;          // 0..31
    const int n = n0 + nl2, c = c0 + cl2;
    if (n < NANCH && c < NCLS)
      scoresT[((size_t)b * NCLS + c) * NANCH + n] = tile[nl2][cl2];
  }
}

// ---------------------------------------------------------------------------
// K2: per (b,c): exact top-300 by radix select (keys in registers), sorted by
// bitonic sort, 300x300 IoU bitmask, greedy NMS scan, write zero-padded rows.
// ---------------------------------------------------------------------------
__global__ __launch_bounds__(256) void k_classnms(const float* __restrict__ in,
                                                  const float* __restrict__ scoresT,
                                                  float* __restrict__ fsc,
                                                  float* __restrict__ frows) {
  const int t  = threadIdx.x;
  const int bc = blockIdx.x;
  const int b  = bc / NCLS;
  const int c  = bc % NCLS;

  __shared__ uint32_t hist[256];
  __shared__ uint32_t sh_sel, sh_kk, sh_cnt, sh_cnteq;
  __shared__ unsigned long long skey[512];
  __shared__ float by0[MM], bx0[MM], by1[MM], bx1[MM], bval[MM];
  __shared__ uint32_t vmaskb[10], keepb[10];
  __shared__ uint32_t sup[MM][10];

  // ---- load masked score keys into registers -------------------------------
  uint32_t key[KPT2];
  const float* sc = scoresT + ((size_t)b * NCLS + c) * NANCH;
#pragma unroll
  for (int j = 0; j < KPT2; ++j) {
    const int n = t + 256 * j;
    float s = (n < NANCH) ? sc[n] : 0.f;
    float m = (s > CONF_T) ? s : NEGV;
    key[j] = (n < NANCH) ? f2ord(m) : 0u;   // pad key 0 < any real key
  }

  // ---- radix select: T = 300th-largest key ---------------------------------
  uint32_t prefix = 0, maskhi = 0;
  if (t == 0) sh_kk = MM;
  __syncthreads();
  for (int pass = 0; pass < 4; ++pass) {
    const int shift = 24 - 8 * pass;
    hist[t] = 0;
    __syncthreads();
#pragma unroll
    for (int j = 0; j < KPT2; ++j) {
      const uint32_t k = key[j];
      if ((k & maskhi) == prefix) atomicAdd(&hist[(k >> shift) & 255u], 1u);
    }
    __syncthreads();
    if (t == 0) {
      uint32_t kk = sh_kk, cum = 0, sel = 0;
      for (int bin = 255; bin >= 0; --bin) {
        const uint32_t h = hist[bin];
        if (cum + h >= kk) { sel = (uint32_t)bin; sh_kk = kk - cum; break; }
        cum += h;
      }
      sh_sel = sel;
    }
    __syncthreads();
    prefix |= sh_sel << shift;
    maskhi |= 0xFFu << shift;
    __syncthreads();
  }
  const uint32_t T    = prefix;
  const uint32_t kkeq = sh_kk;      // how many ==T entries we still need

  // ---- exact tie handling: kkeq smallest indices among key==T --------------
  if (t == 0) sh_cnteq = 0;
  __syncthreads();
  {
    uint32_t ce = 0;
#pragma unroll
    for (int j = 0; j < KPT2; ++j) if (key[j] == T) ++ce;
    if (ce) atomicAdd(&sh_cnteq, ce);
  }
  __syncthreads();
  uint32_t idxT = 0xFFFFFFFFu;
  if (sh_cnteq > kkeq) {            // uniform branch (shared values)
    uint32_t p2 = 0, m2 = 0;
    if (t == 0) sh_kk = kkeq;
    __syncthreads();
    for (int pass = 0; pass < 4; ++pass) {
      const int shift = 24 - 8 * pass;
      hist[t] = 0;
      __syncthreads();
#pragma unroll
      for (int j = 0; j < KPT2; ++j) {
        const uint32_t n = (uint32_t)(t + 256 * j);
        if (key[j] == T && ((n & m2) == p2))
          atomicAdd(&hist[(n >> shift) & 255u], 1u);
      }
      __syncthreads();
      if (t == 0) {
        uint32_t kk = sh_kk, cum = 0, sel = 0;
        for (int bin = 0; bin < 256; ++bin) {      // ascending: smallest index
          const uint32_t h = hist[bin];
          if (cum + h >= kk) { sel = (uint32_t)bin; sh_kk = kk - cum; break; }
          cum += h;
        }
        sh_sel = sel;
      }
      __syncthreads();
      p2 |= sh_sel << shift;
      m2 |= 0xFFu << shift;
      __syncthreads();
    }
    idxT = p2;
  }

  // ---- compaction of exactly 300 (key, idx) pairs --------------------------
  if (t == 0) sh_cnt = 0;
  __syncthreads();
#pragma unroll
  for (int j = 0; j < KPT2; ++j) {
    const uint32_t k = key[j];
    const uint32_t n = (uint32_t)(t + 256 * j);
    const bool take = (n < NANCH) && ((k > T) || (k == T && n <= idxT));
    if (take) {
      const uint32_t pos = atomicAdd(&sh_cnt, 1u);
      if (pos < 512u)
        skey[pos] = ((unsigned long long)k << 32) | (uint32_t)(~n);
    }
  }
  __syncthreads();
  {
    uint32_t cnt = sh_cnt; if (cnt > 512u) cnt = 512u;
    for (int i = t; i < 512; i += 256)
      if ((uint32_t)i >= cnt) skey[i] = 0ull;
  }

  // ---- bitonic sort, descending (score desc, index asc on ties) ------------
  for (int k2 = 2; k2 <= 512; k2 <<= 1)
    for (int j = k2 >> 1; j > 0; j >>= 1) {
      __syncthreads();
      for (int i = t; i < 512; i += 256) {
        const int ij = i ^ j;
        if (ij > i) {
          unsigned long long a = skey[i], bb = skey[ij];
          const bool dir = ((i & k2) == 0);        // descending overall
          if (dir ? (a < bb) : (a > bb)) { skey[i] = bb; skey[ij] = a; }
        }
      }
    }
  if (t < 10) { vmaskb[t] = 0u; }
  __syncthreads();

  // ---- gather candidate boxes (center -> corner) ---------------------------
  for (int i = t; i < MM; i += 256) {
    const unsigned long long e = skey[i];
    if (e) {
      const uint32_t k = (uint32_t)(e >> 32);
      const uint32_t n = ~((uint32_t)e);
      const float v = ord2f(k);
      const float4 bx = *(const float4*)(in + ((size_t)b * NANCH + n) * 604 + 600);
      const float cx = bx.x, cy = bx.y, w = bx.z, h = bx.w;
      by0[i] = cy - h * 0.5f; bx0[i] = cx - w * 0.5f;
      by1[i] = cy + h * 0.5f; bx1[i] = cx + w * 0.5f;
      bval[i] = v;
      if (v > CONF_T) atomicOr(&vmaskb[i >> 5], 1u << (i & 31));
    } else {
      by0[i] = bx0[i] = by1[i] = bx1[i] = 0.f;
      bval[i] = NEGV;
    }
  }
  __syncthreads();

  // ---- suppression bitmask rows: sup[i] bit j <=> iou(i,j) > thr -----------
  for (int i = t; i < MM; i += 256) {
    const float y0i = by0[i], x0i = bx0[i], y1i = by1[i], x1i = bx1[i];
    const float ai = (y1i - y0i) * (x1i - x0i);
    uint32_t row[10];
#pragma unroll
    for (int q = 0; q < 10; ++q) row[q] = 0u;
    for (int j = 0; j < MM; ++j) {
      float ih = fminf(y1i, by1[j]) - fmaxf(y0i, by0[j]); ih = fmaxf(ih, 0.f);
      float iw = fminf(x1i, bx1[j]) - fmaxf(x0i, bx0[j]); iw = fmaxf(iw, 0.f);
      const float inter = ih * iw;
      const float aj = (by1[j] - by0[j]) * (bx1[j] - bx0[j]);
      const float un = ai + aj - inter;
      const float iou = (un > 0.f) ? (inter / un) : 0.f;
      if (j != i && iou > IOU_T) row[j >> 5] |= (1u << (j & 31));
    }
#pragma unroll
    for (int q = 0; q < 10; ++q) sup[i][q] = row[q];
  }
  __syncthreads();

  // ---- greedy NMS scan (single lane, 300 steps) ----------------------------
  if (t == 0) {
    uint32_t acc[10] = {0}, km[10] = {0};
    for (int i = 0; i < MM; ++i) {
      const int w2 = i >> 5, bb = i & 31;
      if (((vmaskb[w2] >> bb) & 1u) && !((acc[w2] >> bb) & 1u)) {
        km[w2] |= 1u << bb;
#pragma unroll
        for (int q = 0; q < 10; ++q) acc[q] |= sup[i][q];
      }
    }
#pragma unroll
    for (int q = 0; q < 10; ++q) keepb[q] = km[q];
  }
  __syncthreads();

  // ---- emit zero-padded rows -----------------------------------------------
  float* fs = fsc + (size_t)b * NTOT;
  float* fr = frows + (size_t)b * NTOT * 6;
  const float clsid = (float)(c + 1);
  for (int i = t; i < MM; i += 256) {
    const bool kp = (keepb[i >> 5] >> (i & 31)) & 1u;
    const size_t o = (size_t)c * MM + i;
    const float v = kp ? bval[i] : 0.f;
    fs[o] = v;
    float* r = fr + o * 6;
    r[0] = kp ? clsid  : 0.f;
    r[1] = v;
    r[2] = kp ? by0[i] : 0.f;
    r[3] = kp ? bx0[i] : 0.f;
    r[4] = kp ? by1[i] : 0.f;
    r[5] = kp ? bx1[i] : 0.f;
  }
}

// ---------------------------------------------------------------------------
// K3: per image, exact top-200 of 179700 row scores (radix select with exact
// lower-index tie handling), sort, gather, corner -> center, write output.
// ---------------------------------------------------------------------------
__global__ __launch_bounds__(1024) void k_topk(const float* __restrict__ fsc,
                                               const float* __restrict__ frows,
                                               float* __restrict__ out) {
  const int t = threadIdx.x;
  const int b = blockIdx.x;
  const float* fs = fsc + (size_t)b * NTOT;
  const float* fr = frows + (size_t)b * NTOT * 6;

  __shared__ uint32_t hist[256];
  __shared__ uint32_t sh_sel, sh_kk, sh_cnt, sh_cnteq;
  __shared__ unsigned long long skey[256];

  uint32_t prefix = 0, maskhi = 0;
  if (t == 0) sh_kk = KOUT;
  __syncthreads();
  for (int pass = 0; pass < 4; ++pass) {
    const int shift = 24 - 8 * pass;
    if (t < 256) hist[t] = 0;
    __syncthreads();
    for (int n = t; n < NTOT; n += 1024) {
      const uint32_t k = f2ord(fs[n]);
      if ((k & maskhi) == prefix) atomicAdd(&hist[(k >> shift) & 255u], 1u);
    }
    __syncthreads();
    if (t == 0) {
      uint32_t kk = sh_kk, cum = 0, sel = 0;
      for (int bin = 255; bin >= 0; --bin) {
        const uint32_t h = hist[bin];
        if (cum + h >= kk) { sel = (uint32_t)bin; sh_kk = kk - cum; break; }
        cum += h;
      }
      sh_sel = sel;
    }
    __syncthreads();
    prefix |= sh_sel << shift;
    maskhi |= 0xFFu << shift;
    __syncthreads();
  }
  const uint32_t T    = prefix;
  const uint32_t kkeq = sh_kk;

  if (t == 0) sh_cnteq = 0;
  __syncthreads();
  {
    uint32_t ce = 0;
    for (int n = t; n < NTOT; n += 1024) if (f2ord(fs[n]) == T) ++ce;
    if (ce) atomicAdd(&sh_cnteq, ce);
  }
  __syncthreads();
  uint32_t idxT = 0xFFFFFFFFu;
  if (sh_cnteq > kkeq) {            // big tie class (e.g. score==0): exact
    uint32_t p2 = 0, m2 = 0;
    if (t == 0) sh_kk = kkeq;
    __syncthreads();
    for (int pass = 0; pass < 4; ++pass) {
      const int shift = 24 - 8 * pass;
      if (t < 256) hist[t] = 0;
      __syncthreads();
      for (int n = t; n < NTOT; n += 1024) {
        if (f2ord(fs[n]) == T && (((uint32_t)n & m2) == p2))
          atomicAdd(&hist[((uint32_t)n >> shift) & 255u], 1u);
      }
      __syncthreads();
      if (t == 0) {
        uint32_t kk = sh_kk, cum = 0, sel = 0;
        for (int bin = 0; bin < 256; ++bin) {
          const uint32_t h = hist[bin];
          if (cum + h >= kk) { sel = (uint32_t)bin; sh_kk = kk - cum; break; }
          cum += h;
        }
        sh_sel = sel;
      }
      __syncthreads();
      p2 |= sh_sel << shift;
      m2 |= 0xFFu << shift;
      __syncthreads();
    }
    idxT = p2;
  }

  if (t == 0) sh_cnt = 0;
  __syncthreads();
  for (int n = t; n < NTOT; n += 1024) {
    const uint32_t k = f2ord(fs[n]);
    const bool take = (k > T) || (k == T && (uint32_t)n <= idxT);
    if (take) {
      const uint32_t pos = atomicAdd(&sh_cnt, 1u);
      if (pos < 256u)
        skey[pos] = ((unsigned long long)k << 32) | (uint32_t)(~(uint32_t)n);
    }
  }
  __syncthreads();
  {
    uint32_t cnt = sh_cnt; if (cnt > 256u) cnt = 256u;
    if (t < 256 && (uint32_t)t >= cnt) skey[t] = 0ull;
  }

  for (int k2 = 2; k2 <= 256; k2 <<= 1)
    for (int j = k2 >> 1; j > 0; j >>= 1) {
      __syncthreads();
      if (t < 256) {
        const int i = t, ij = i ^ j;
        if (ij > i) {
          unsigned long long a = skey[i], bb = skey[ij];
          const bool dir = ((i & k2) == 0);
          if (dir ? (a < bb) : (a > bb)) { skey[i] = bb; skey[ij] = a; }
        }
      }
    }
  __syncthreads();

  if (t < KOUT) {
    const unsigned long long e = skey[t];
    float r0 = 0.f, r1 = 0.f, r2 = 0.f, r3 = 0.f, r4 = 0.f, r5 = 0.f;
    if (e) {
      const uint32_t n = ~((uint32_t)e);
      const float* r = fr + (size_t)n * 6;
      r0 = r[0]; r1 = r[1]; r2 = r[2]; r3 = r[3]; r4 = r[4]; r5 = r[5];
    }
    const float th = r4 - r2;
    const float tw = r5 - r3;
    float* o = out + ((size_t)b * KOUT + t) * 6;
    o[0] = r0;
    o[1] = r1;
    o[2] = r3 + tw * 0.5f;   // cx
    o[3] = r2 + th * 0.5f;   // cy
    o[4] = tw;               // w
    o[5] = th;               // h
  }
}

// ---------------------------------------------------------------------------
extern "C" void kernel_launch(void* const* d_in, const int* in_sizes, int n_in,
                              void* d_out, int out_size, void* d_ws, size_t ws_size,
                              hipStream_t stream) {
  (void)in_sizes; (void)n_in; (void)out_size; (void)ws_size;
  const float* in = (const float*)d_in[0];
  float* ws      = (float*)d_ws;
  float* scoresT = ws;                                         // B*C*N   floats
  float* fsc     = scoresT + (size_t)BATCH * NCLS * NANCH;     // B*C*M   floats
  float* frows   = fsc + (size_t)BATCH * NTOT;                 // B*C*M*6 floats

  dim3 g1((NANCH + 63) / 64, (NCLS + 31) / 32, BATCH);
  k_transpose<<<g1, 256, 0, stream>>>(in, scoresT);
  k_classnms<<<dim3(BATCH * NCLS), 256, 0, stream>>>(in, scoresT, fsc, frows);
  k_topk<<<dim3(BATCH), 1024, 0, stream>>>(fsc, frows, (float*)d_out);
}